// VectorQuantizerRandomRestart_22900765622860
// MI455X (gfx1250) — compile-verified
//
#include <hip/hip_runtime.h>
#include <math.h>

// ---- problem constants (from reference) ----
#define NE     1024         // number of codes
#define ED     256          // embedding dim
#define BT     32           // batch
#define TT     1024         // tokens per batch
#define NROW   32768        // BT*TT rows of zf
#define ZSZ    8388608      // BT*ED*TT
#define MUC    0.8f

// ---- output layout (floats, concatenated tuple) ----
#define O_LOSS      0
#define O_ZQ        1
#define O_PERP      8388609
#define O_ONEHOT    8388610
#define O_IDX       41943042
#define O_NEWW      41975810
#define O_NEWKSUM   42237954
#define O_NEWKELEM  42500098

// ---- workspace layout (float units) ----
#define W_LOSS    0
#define W_COUNTS  64        // 1024 floats
#define W_ENORM   2048      // 1024 floats
#define W_KSUM    4096      // 262144 floats
#define W_IDX     266240    // 32768 ints (int view)

#define BPAD   260          // padded B row stride (260 mod 64 = 4 -> conflict-free WMMA reads)
#define BBUF   4160         // 16 * BPAD floats per B buffer

typedef float v2f __attribute__((ext_vector_type(2)));
typedef float v8f __attribute__((ext_vector_type(8)));

// CDNA5 async global->LDS copy (ASYNCcnt-tracked, in-order completion, no VGPR staging)
__device__ __forceinline__ void async_b128(const float* gaddr, const float* lds) {
    asm volatile("global_load_async_to_lds_b128 %0, %1, off"
                 :: "v"((unsigned)(size_t)lds),
                    "v"((unsigned long long)(size_t)gaddr)
                 : "memory");
}

// K0: zero accumulators, precompute ||e_j||^2
__global__ __launch_bounds__(256) void k_init(const float* __restrict__ emb,
                                              float* __restrict__ ws) {
    int i = blockIdx.x * blockDim.x + threadIdx.x;
    for (int p = i; p < NE * ED; p += gridDim.x * blockDim.x)
        ws[W_KSUM + p] = 0.0f;
    if (i < NE) {
        ws[W_COUNTS + i] = 0.0f;
        const float* e = emb + (size_t)i * ED;
        float s = 0.0f;
        #pragma unroll 4
        for (int k = 0; k < ED; ++k) s += e[k] * e[k];
        ws[W_ENORM + i] = s;
    }
    if (i == 0) ws[W_LOSS] = 0.0f;
}

// K1: distances via V_WMMA_F32_16X16X4_F32 + per-row argmin.
// Block: 256 threads (8 waves), owns 128 rows; loops all 64 code tiles.
// A (8*16x256 f32, k-major) and double-buffered B (16x256+pad) are staged with
// GLOBAL_LOAD_ASYNC_TO_LDS_B128; B tile nt+1 streams while WMMAs consume tile nt.
__global__ __launch_bounds__(256) void k_dist_argmin(const float* __restrict__ z,
                                                     const float* __restrict__ emb,
                                                     const float* __restrict__ ws,
                                                     int* __restrict__ widx) {
    extern __shared__ float smem[];
    float* ldsA = smem;                       // 32768 floats
    float* ldsB = smem + 32768;               // 2 * 4160 floats (double buffer)
    float* ldsZ = smem + 32768 + 2 * BBUF;    // 128 floats

    const float* enorm = ws + W_ENORM;
    const int tid  = threadIdx.x;
    const int r0   = blockIdx.x * 128;        // 128 rows, always within one batch b
    const int b    = r0 >> 10;
    const int t0   = r0 & 1023;
    const float* zb = z + (size_t)b * (ED * TT);

    // async-stage A: zf rows (transposed from z[b,k,t]) -> ldsA[(ml>>4)*4096 + k*16 + (ml&15)]
    // 8192 x b128; LDS dest stays 16B-contiguous because ml&15 in {0,4,8,12}
    for (int i = 0; i < 32; ++i) {
        int f4 = i * 256 + tid;
        int k  = f4 >> 5;                     // 0..255
        int ml = (f4 & 31) * 4;               // local row, multiple of 4
        async_b128(zb + k * TT + t0 + ml,
                   ldsA + (ml >> 4) * 4096 + k * 16 + (ml & 15));
    }
    // async-stage B tile 0 into buffer 0: ldsB[n*BPAD + k] = emb[n0+n][k]
    const int nB   = tid >> 4;                // 0..15: code row within tile
    const int kq16 = tid & 15;
    {
        const float* ep = emb + (size_t)nB * ED;
        float* bp = ldsB + nB * BPAD;
        #pragma unroll
        for (int q = 0; q < 4; ++q) {
            int kq = q * 16 + kq16;           // qword index 0..63
            async_b128(ep + kq * 4, bp + kq * 4);
        }
    }
    asm volatile("s_wait_asynccnt 0x4" ::: "memory");   // A's 32 done (in-order); B tile0 may trail
    __syncthreads();

    if (tid < 128) {                          // znorm per local row
        const float* ap = ldsA + (tid >> 4) * 4096 + (tid & 15);
        float s = 0.0f;
        #pragma unroll 4
        for (int k = 0; k < ED; ++k) { float v = ap[k * 16]; s += v * v; }
        ldsZ[tid] = s;
    }
    __syncthreads();

    const int w    = tid >> 5;
    const int lane = tid & 31;
    const int half = lane >> 4;
    const int nl   = lane & 15;
    const float* Aw = ldsA + w * 4096;
    const int koff  = half * 2;               // ISA: lanes 0-15 hold K=0,1; lanes 16-31 hold K=2,3

    float bestv[8]; int besti[8];
    #pragma unroll
    for (int v = 0; v < 8; ++v) { bestv[v] = 3.4e38f; besti[v] = 0; }

    for (int nt = 0; nt < 64; ++nt) {
        // kick off next B tile into the other buffer while this one is consumed
        if (nt < 63) {
            const float* ep = emb + (size_t)((nt + 1) * 16 + nB) * ED;
            float* bp = ldsB + ((nt + 1) & 1) * BBUF + nB * BPAD;
            #pragma unroll
            for (int q = 0; q < 4; ++q) {
                int kq = q * 16 + kq16;
                async_b128(ep + kq * 4, bp + kq * 4);
            }
            asm volatile("s_wait_asynccnt 0x4" ::: "memory");  // current tile's 4 landed
        } else {
            asm volatile("s_wait_asynccnt 0x0" ::: "memory");
        }
        __syncthreads();                      // all waves' async writes visible

        const float* Bc = ldsB + (nt & 1) * BBUF;
        v8f acc = {0.f,0.f,0.f,0.f,0.f,0.f,0.f,0.f};
        #pragma unroll 8
        for (int kb = 0; kb < 64; ++kb) {
            int k0 = kb * 4 + koff;
            v2f a, bb;
            a.x  = Aw[k0 * 16 + nl];
            a.y  = Aw[(k0 + 1) * 16 + nl];
            bb.x = Bc[nl * BPAD + k0];        // B[k][n] = emb[n][k]; row-padded, bank-clean
            bb.y = Bc[nl * BPAD + k0 + 1];
            acc = __builtin_amdgcn_wmma_f32_16x16x4_f32(
                    false, a, false, bb, (short)0, acc, false, false);
        }

        const int n0 = nt * 16;
        float en = enorm[n0 + nl];
        const int n = n0 + nl;
        #pragma unroll
        for (int v = 0; v < 8; ++v) {
            int m = v + half * 8;                             // C layout: VGPR v -> M=v(+8)
            float d = ldsZ[w * 16 + m] + en - 2.0f * acc[v];
            if (d < bestv[v]) { bestv[v] = d; besti[v] = n; } // strict < keeps first min
        }
        __syncthreads();                      // all waves done reading Bc before it is refilled
    }

    // reduce argmin across the 16 lanes holding each row (width 16: halves independent)
    #pragma unroll
    for (int v = 0; v < 8; ++v) {
        float bv = bestv[v]; int bi = besti[v];
        #pragma unroll
        for (int m = 8; m >= 1; m >>= 1) {
            float ov = __shfl_xor(bv, m, 16);
            int   oi = __shfl_xor(bi, m, 16);
            if (ov < bv || (ov == bv && oi < bi)) { bv = ov; bi = oi; }
        }
        if (nl == 0) {
            int row = r0 + w * 16 + v + half * 8;
            widx[row] = bi;
        }
    }
}

// K2: z_q gather (original (b,k,t) layout) + fused squared-error reduction
__global__ __launch_bounds__(256) void k_gather_loss(const float* __restrict__ z,
                                                     const float* __restrict__ emb,
                                                     const int* __restrict__ widx,
                                                     float* __restrict__ out,
                                                     float* __restrict__ ws) {
    __shared__ float red[256];
    float local = 0.0f;
    int stride = gridDim.x * blockDim.x;
    for (int i = blockIdx.x * blockDim.x + threadIdx.x; i < ZSZ; i += stride) {
        int b = i >> 18;                  // / (ED*TT)
        int rem = i & 262143;
        int k = rem >> 10;
        int t = rem & 1023;
        int j = widx[b * TT + t];
        float q = emb[(size_t)j * ED + k];
        out[(size_t)O_ZQ + i] = q;
        float d = q - z[i];
        local += d * d;
    }
    red[threadIdx.x] = local;
    __syncthreads();
    for (int s = 128; s > 0; s >>= 1) {
        if (threadIdx.x < s) red[threadIdx.x] += red[threadIdx.x + s];
        __syncthreads();
    }
    if (threadIdx.x == 0) atomicAdd(ws + W_LOSS, red[0]);
}

// K3: counts + k_sum scatter (coalesced z reads; scattered L2 atomics)
__global__ __launch_bounds__(256) void k_scatter(const float* __restrict__ z,
                                                 const int* __restrict__ widx,
                                                 float* __restrict__ ws) {
    int stride = gridDim.x * blockDim.x;
    int tid0 = blockIdx.x * blockDim.x + threadIdx.x;
    for (int i = tid0; i < ZSZ; i += stride) {
        int k = i >> 15;                 // / NROW
        int r = i & 32767;
        int b = r >> 10, t = r & 1023;
        float v = z[(size_t)b * (ED * TT) + k * TT + t];
        atomicAdd(&ws[W_KSUM + widx[r] * ED + k], v);
    }
    for (int r = tid0; r < NROW; r += stride)
        atomicAdd(&ws[W_COUNTS + widx[r]], 1.0f);
}

// K4: onehot (128 MB stream) + idx-as-float
__global__ __launch_bounds__(256) void k_onehot(const int* __restrict__ widx,
                                                float* __restrict__ out) {
    int stride = gridDim.x * blockDim.x;
    int tid0 = blockIdx.x * blockDim.x + threadIdx.x;
    for (long i = tid0; i < (long)NROW * NE; i += stride) {
        int r = (int)(i >> 10);
        int c = (int)(i & 1023);
        out[(size_t)O_ONEHOT + i] = (c == widx[r]) ? 1.0f : 0.0f;
    }
    for (int r = tid0; r < NROW; r += stride)
        out[(size_t)O_IDX + r] = (float)widx[r];
}

// K5: per-code EMA update + random-restart rows
__global__ __launch_bounds__(256) void k_update(const float* __restrict__ z,
                                                const float* __restrict__ ksum_in,
                                                const float* __restrict__ kelem_in,
                                                const float* __restrict__ ws,
                                                float* __restrict__ out) {
    int j = blockIdx.x;
    int k = threadIdx.x;
    float cnt = ws[W_COUNTS + j];
    float nke = MUC * kelem_in[j] + (1.0f - MUC) * cnt;
    float nks = MUC * ksum_in[(size_t)j * ED + k] + (1.0f - MUC) * ws[W_KSUM + j * ED + k];
    out[(size_t)O_NEWKSUM + (size_t)j * ED + k] = nks;
    // deterministic injective stand-in for jax permutation(key 42)
    unsigned p = (2654435761u * (unsigned)j + 40503u) & 32767u;
    float kr = z[(size_t)(p >> 10) * (ED * TT) + (size_t)k * TT + (p & 1023)];
    bool use = nke >= 1.0f;
    out[(size_t)O_NEWW + (size_t)j * ED + k] = use ? (nks / nke) : kr;
    if (k == 0) out[(size_t)O_NEWKELEM + j] = nke;
}

// K6: loss + perplexity scalars
__global__ __launch_bounds__(256) void k_scalars(const float* __restrict__ ws,
                                                 float* __restrict__ out) {
    __shared__ float red[256];
    float s = 0.0f;
    for (int i = threadIdx.x; i < NE; i += 256) {
        float e = ws[W_COUNTS + i] * (1.0f / (float)NROW);
        s += e * logf(e + 1e-10f);
    }
    red[threadIdx.x] = s;
    __syncthreads();
    for (int st = 128; st > 0; st >>= 1) {
        if (threadIdx.x < st) red[threadIdx.x] += red[threadIdx.x + st];
        __syncthreads();
    }
    if (threadIdx.x == 0) {
        out[O_PERP] = expf(-red[0]);
        out[O_LOSS] = 1.25f * ws[W_LOSS] / (float)ZSZ;   // (1+BETA)*MSE
    }
}

extern "C" void kernel_launch(void* const* d_in, const int* in_sizes, int n_in,
                              void* d_out, int out_size, void* d_ws, size_t ws_size,
                              hipStream_t stream) {
    const float* z     = (const float*)d_in[0];
    const float* emb   = (const float*)d_in[1];
    const float* ksum  = (const float*)d_in[2];
    const float* kelem = (const float*)d_in[3];
    float* out = (float*)d_out;
    float* ws  = (float*)d_ws;
    int*   widx = (int*)(ws + W_IDX);

    constexpr size_t SMEM = (32768 + 2 * BBUF + 128) * sizeof(float);  // 164864 B (<320KB WGP LDS)
    hipFuncSetAttribute(reinterpret_cast<const void*>(k_dist_argmin),
                        hipFuncAttributeMaxDynamicSharedMemorySize, (int)SMEM);

    k_init<<<1024, 256, 0, stream>>>(emb, ws);
    k_dist_argmin<<<256, 256, SMEM, stream>>>(z, emb, ws, widx);
    k_gather_loss<<<2048, 256, 0, stream>>>(z, emb, widx, out, ws);
    k_scatter<<<4096, 256, 0, stream>>>(z, widx, ws);
    k_onehot<<<8192, 256, 0, stream>>>(widx, out);
    k_update<<<NE, ED, 0, stream>>>(z, ksum, kelem, ws, out);
    k_scalars<<<1, 256, 0, stream>>>(ws, out);
}